// PGONet_51599737094645
// MI455X (gfx1250) — compile-verified
//
#include <hip/hip_runtime.h>
#include <hip/hip_bf16.h>
#include <stdint.h>

// ---------------------------------------------------------------------------
// PGONet wave-step, MI455X (gfx1250). Bandwidth-bound (~200MB @ 23.3TB/s
// => ~8.6us floor). CDNA5 async global->LDS DMA (ASYNCcnt) stages the
// stencil tiles; interior tiles use b128 async loads (512B/wave/issue);
// outputs stream out via non-temporal b128 stores (don't thrash 192MB L2).
// ---------------------------------------------------------------------------

#define H    2048
#define DT   0.001f
#define DX   0.02f
static constexpr size_t HH = (size_t)H * (size_t)H;

typedef float v4f __attribute__((ext_vector_type(4)));

// LDS input tile: rows 36 (halo 2 top/bottom), cols 40 (halo 4 left / 4 right,
// widened left halo keeps b128 global/LDS addresses 16B-aligned).
#define SROWS 36
#define SCOLS 40
// input LDS (li,lj)  <->  global (i0-2+li, j0-4+lj)

// ---- CDNA5 async global->LDS (GLOBAL_LOAD_ASYNC_TO_LDS_*, GV mode) --------
__device__ __forceinline__ unsigned lds_off_of(const void* p) {
    // generic ptr to LDS = {aperture_hi, lds_offset_lo32}; truncate => LDS addr
    return (unsigned)(uintptr_t)p;
}
__device__ __forceinline__ void async_load_b32_to_lds(unsigned lds_addr, const float* gptr) {
    asm volatile("global_load_async_to_lds_b32 %0, %1, off"
                 :
                 : "v"(lds_addr), "v"((unsigned long long)(uintptr_t)gptr)
                 : "memory");
}
__device__ __forceinline__ void async_load_b128_to_lds(unsigned lds_addr, const float* gptr) {
    asm volatile("global_load_async_to_lds_b128 %0, %1, off"
                 :
                 : "v"(lds_addr), "v"((unsigned long long)(uintptr_t)gptr)
                 : "memory");
}
__device__ __forceinline__ void wait_asynccnt0() {
    asm volatile("s_wait_asynccnt 0" ::: "memory");
}

// ---------------------------------------------------------------------------
// Main fused kernel: weight-norm 3x3x3 conv + both 5-pt stencils + BCs.
// 32x32 output tile per 256-thread block (8 wave32s).
// ---------------------------------------------------------------------------
__global__ __launch_bounds__(256)
void pgonet_main(const float* __restrict__ speed,
                 const float* __restrict__ batch,
                 const float* __restrict__ conv_v,
                 const float* __restrict__ conv_g,
                 const float* __restrict__ conv_b,
                 const int*   __restrict__ bsize_p,
                 const int*   __restrict__ flag_p,
                 const int*   __restrict__ flagnum_p,
                 float* __restrict__ out)
{
    __shared__ __align__(16) float s_in[3][SROWS * SCOLS];
    __shared__ float s_y[34 * 35];
    __shared__ float s_w[28];

    const int tid = threadIdx.x;
    const int i0  = blockIdx.y * 32;
    const int j0  = blockIdx.x * 32;

    const int ntb  = flag_p[0] - 1;
    const int step = flagnum_p[0] - 1;
    const int bs   = bsize_p[0];
    const float* xtt_g = batch + (size_t)(ntb * bs + step) * HH;
    const float* xt_g  = xtt_g + HH;

    const float* planes[3] = { xtt_g, xt_g, speed };

    // ---- stage 1: async-DMA tile+halo of {xtt, xt, speed} into LDS --------
    // Fast path (interior tiles, 93.8%): rows i0-2..i0+33, cols j0-4..j0+35
    // all in range -> b128 async loads, 16B-aligned.
    const bool interior_tile = (i0 >= 32) && (i0 <= H - 64) &&
                               (j0 >= 32) && (j0 <= H - 64);
    if (interior_tile) {
        // 3 planes * 36 rows * 10 float4s = 1080 b128 transfers
        for (int idx = tid; idx < 3 * SROWS * (SCOLS / 4); idx += 256) {
            const int p   = idx / (SROWS * (SCOLS / 4));
            const int rem = idx % (SROWS * (SCOLS / 4));
            const int li  = rem / (SCOLS / 4);
            const int ljv = rem % (SCOLS / 4);            // float4 index in row
            const float* src = planes[p] +
                (size_t)(i0 - 2 + li) * H + (size_t)(j0 - 4 + ljv * 4);
            async_load_b128_to_lds(lds_off_of(&s_in[p][li * SCOLS + ljv * 4]), src);
        }
    } else {
        // Boundary tiles: per-element b32 with zero-fill (= conv zero padding)
        #pragma unroll
        for (int p = 0; p < 3; ++p) {
            for (int idx = tid; idx < SROWS * SCOLS; idx += 256) {
                const int li = idx / SCOLS, lj = idx % SCOLS;
                const int gi = i0 - 2 + li, gj = j0 - 4 + lj;
                float* dst = &s_in[p][li * SCOLS + lj];
                if (gi >= 0 && gi < H && gj >= 0 && gj < H) {
                    async_load_b32_to_lds(lds_off_of(dst),
                                          planes[p] + (size_t)gi * H + gj);
                } else {
                    *dst = 0.0f;
                }
            }
        }
    }
    // weight normalization (uniform; overlapped with the async DMA)
    if (tid == 0) {
        float ss = 0.0f;
        #pragma unroll
        for (int k = 0; k < 27; ++k) ss += conv_v[k] * conv_v[k];
        const float sc = conv_g[0] / sqrtf(ss);
        #pragma unroll
        for (int k = 0; k < 27; ++k) s_w[k] = conv_v[k] * sc;
        s_w[27] = conv_b[0];
    }
    wait_asynccnt0();
    __syncthreads();

    // ---- stage 2: y = conv3x3x3 + b on 34x34 (tile + halo 1) --------------
    // y (li',lj') <-> global (i0-1+li', j0-1+lj');
    // input index: row li'+di, col lj'+dj+2 (di,dj in 0..2)
    for (int idx = tid; idx < 34 * 34; idx += 256) {
        const int li = idx / 34, lj = idx % 34;
        float acc = s_w[27];
        #pragma unroll
        for (int c = 0; c < 3; ++c)
            #pragma unroll
            for (int di = 0; di < 3; ++di)
                #pragma unroll
                for (int dj = 0; dj < 3; ++dj)
                    acc = fmaf(s_w[c * 9 + di * 3 + dj],
                               s_in[c][(li + di) * SCOLS + (lj + dj + 2)], acc);
        s_y[li * 35 + lj] = acc;
    }
    __syncthreads();

    // ---- stage 3: stencils + boundary conditions, NT b128 stores ----------
    const int   ty    = tid >> 3;        // 0..31 : row in tile
    const int   tx0   = (tid & 7) * 4;   // 0..28 : first of 4 cols
    const int   i     = i0 + ty;
    const float rcoef = (DT * DT) / (DX * DX);
    const float rdx   = DT / DX;

    v4f o2, o5, o6;
    #pragma unroll
    for (int k = 0; k < 4; ++k) {
        const int j = j0 + tx0 + k;
        const int a = ty + 2, b = tx0 + k + 4;        // s_in center
        const float xt_c  = s_in[1][a * SCOLS + b];
        const float xt_n  = s_in[1][(a - 1) * SCOLS + b];
        const float xt_s  = s_in[1][(a + 1) * SCOLS + b];
        const float xt_w  = s_in[1][a * SCOLS + b - 1];
        const float xt_e  = s_in[1][a * SCOLS + b + 1];
        const float xtt_c = s_in[0][a * SCOLS + b];
        const float sp_c  = s_in[2][a * SCOLS + b];
        const int ya = ty + 1, yb = tx0 + k + 1;      // s_y center
        const float y_c = s_y[ya * 35 + yb];
        const float y_n = s_y[(ya - 1) * 35 + yb];
        const float y_s = s_y[(ya + 1) * 35 + yb];
        const float y_w = s_y[ya * 35 + yb - 1];
        const float y_e = s_y[ya * 35 + yb + 1];

        const float coef   = sp_c * sp_c * rcoef;
        const bool  inter  = (i >= 1) && (i <= H - 2) && (j >= 1) && (j <= H - 2);
        const float lap_xt = xt_n + xt_s + xt_w + xt_e - 4.0f * xt_c;
        const float lap_y  = y_n + y_s + y_w + y_e - 4.0f * y_c;
        const float base   = 2.0f * xt_c - xtt_c;

        const float xt4 = inter ? (base + lap_xt * coef) : 0.0f;
        const float x7p = inter ? (base + lap_y  * coef) : y_c;

        // BC precedence per reference .at[].set order:
        // row0 -> leftcol -> diagonal -> rightcol -> rowH-1
        float x7;
        if (i == H - 1)                      x7 = 0.0f;
        else if (j == H - 1)                 x7 = xt_c - rdx * sp_c * (xt_c - xt_w);
        else if (i >= 1 && (i + j == H))     x7 = s_in[1][(a - 1) * SCOLS + (b - 1)];
        else if (j == 0)                     x7 = xt_c - rdx * sp_c * (xt_c - xt_e);
        else if (i == 0)                     x7 = 0.0f;
        else                                 x7 = x7p;

        o2[k] = x7; o5[k] = x7p; o6[k] = xt4;
    }
    const size_t off = (size_t)i * H + (size_t)(j0 + tx0);
    __builtin_nontemporal_store(o2, (v4f*)(out + 2 * HH + off));
    __builtin_nontemporal_store(o5, (v4f*)(out + 5 * HH + off));
    __builtin_nontemporal_store(o6, (v4f*)(out + 6 * HH + off));
}

// ---------------------------------------------------------------------------
// Streaming copy of x_tt -> planes {0,3}, x_t -> planes {1,4}. NT b128.
// ---------------------------------------------------------------------------
__global__ __launch_bounds__(256)
void pgonet_copy(const float* __restrict__ x_tt,
                 const float* __restrict__ x_t,
                 float* __restrict__ out)
{
    const size_t idx = (size_t)blockIdx.x * 256 + threadIdx.x;   // in float4s
    const size_t q   = HH / 4;
    if (idx >= q) return;
    const v4f va = __builtin_nontemporal_load((const v4f*)x_tt + idx);
    const v4f vb = __builtin_nontemporal_load((const v4f*)x_t  + idx);
    v4f* o = (v4f*)out;
    __builtin_nontemporal_store(va, o + 0 * q + idx);
    __builtin_nontemporal_store(vb, o + 1 * q + idx);
    __builtin_nontemporal_store(va, o + 3 * q + idx);
    __builtin_nontemporal_store(vb, o + 4 * q + idx);
}

// ---------------------------------------------------------------------------
// 256-point observation scatter into plane 2 (runs after pgonet_main).
// ---------------------------------------------------------------------------
__global__ void pgonet_scatter(const float* __restrict__ ref_sol,
                               const int* __restrict__ loc_x,
                               const int* __restrict__ loc_y,
                               const int* __restrict__ bsize_p,
                               const int* __restrict__ id_p,
                               const int* __restrict__ flagnum_p,
                               float* __restrict__ out,
                               int n)
{
    const int p = blockIdx.x * blockDim.x + threadIdx.x;
    if (p >= n) return;
    const int step  = flagnum_p[0] - 1;
    const int plane = id_p[0] * bsize_p[0] + step + 2;
    const int x = loc_x[p];
    if (x != -1) {
        const size_t o = (size_t)x * H + (size_t)loc_y[p];
        out[2 * HH + o] = ref_sol[(size_t)plane * HH + o];
    }
}

extern "C" void kernel_launch(void* const* d_in, const int* in_sizes, int n_in,
                              void* d_out, int out_size, void* d_ws, size_t ws_size,
                              hipStream_t stream) {
    const float* ref_speed = (const float*)d_in[0];
    const float* batch     = (const float*)d_in[1];
    const float* x_tt      = (const float*)d_in[2];
    const float* x_t       = (const float*)d_in[3];
    const float* ref_sol   = (const float*)d_in[4];
    const float* conv_v    = (const float*)d_in[5];
    const float* conv_g    = (const float*)d_in[6];
    const float* conv_b    = (const float*)d_in[7];
    const int*   loc_x     = (const int*)d_in[8];
    const int*   loc_y     = (const int*)d_in[9];
    const int*   bsize_p   = (const int*)d_in[10];
    const int*   id_p      = (const int*)d_in[11];
    const int*   flag_p    = (const int*)d_in[12];
    const int*   flagnum_p = (const int*)d_in[13];
    float* out = (float*)d_out;

    dim3 grid(H / 32, H / 32);
    pgonet_main<<<grid, 256, 0, stream>>>(ref_speed, batch, conv_v, conv_g, conv_b,
                                          bsize_p, flag_p, flagnum_p, out);

    const int copyBlocks = (int)((HH / 4 + 255) / 256);
    pgonet_copy<<<copyBlocks, 256, 0, stream>>>(x_tt, x_t, out);

    const int npts = in_sizes[8];
    pgonet_scatter<<<(npts + 255) / 256, 256, 0, stream>>>(ref_sol, loc_x, loc_y,
                                                           bsize_p, id_p, flagnum_p,
                                                           out, npts);
}